// OurModel_57234734187133
// MI455X (gfx1250) — compile-verified
//
#include <hip/hip_runtime.h>

#define H 256

typedef __attribute__((ext_vector_type(2))) float v2f;
typedef __attribute__((ext_vector_type(8))) float v8f;

// -------- tiny prep kernels (negligible cost) --------

// Wc[n][k] = sum_t W_n2e[n][t] * W_in[t][k]   (fused input2hidden + n2e, row-major)
__global__ void combine_weights_kernel(const float* __restrict__ W_in,
                                       const float* __restrict__ W_n2e,
                                       float* __restrict__ Wc) {
    int n = blockIdx.x;
    int k = threadIdx.x;
    float acc = 0.f;
    for (int t = 0; t < H; ++t)
        acc += W_n2e[n * H + t] * W_in[t * H + k];   // scalar bcast * coalesced
    Wc[n * H + k] = acc;
}

// bc[n] = sum_t W_n2e[n][t] * b_in[t] + b_n2e[n]
__global__ void combine_bias_kernel(const float* __restrict__ W_n2e,
                                    const float* __restrict__ b_in,
                                    const float* __restrict__ b_n2e,
                                    float* __restrict__ bc) {
    int n = threadIdx.x;
    float acc = b_n2e[n];
    for (int t = 0; t < H; ++t)
        acc += W_n2e[n * H + t] * b_in[t];
    bc[n] = acc;
}

// Pack row-major weight W[n][k] (used as B[k][n] = W[n][k]) into WMMA-fragment order.
// Bq index: (((ks*2 + h)*8 + t)*16 + lr), float4 value:
//   { B[4ks+2h][32t+lr], B[4ks+2h+1][32t+lr], B[4ks+2h][32t+16+lr], B[4ks+2h+1][32t+16+lr] }
// so lane (h,lr) of the wave owning col-tile t loads one contiguous b128 per k-step,
// and each 16-lane half reads 256 contiguous bytes.
__global__ void pack_b_kernel(const float* __restrict__ W,  // [H][H] row-major (n,k)
                              float4* __restrict__ Bq) {
    int i  = blockIdx.x * blockDim.x + threadIdx.x;  // 0 .. 16383
    int lr = i & 15;
    int t  = (i >> 4) & 7;
    int h  = (i >> 7) & 1;
    int ks = i >> 8;
    int kk = 4 * ks + 2 * h;
    int n  = 32 * t + lr;
    float4 v;
    v.x = W[(n)      * H + kk];
    v.y = W[(n)      * H + kk + 1];
    v.z = W[(n + 16) * H + kk];
    v.w = W[(n + 16) * H + kk + 1];
    Bq[i] = v;
}

__global__ void zero_f32_kernel(float* __restrict__ p, long n) {
    long i = (long)blockIdx.x * blockDim.x + threadIdx.x;
    if (i < n) p[i] = 0.f;
}

// -------- fp32 WMMA GEMM: Y[rows,256] = X[rows,256] @ W^T + bias --------
// Bq = pre-swizzled W (fragment order, see pack_b_kernel). rows % 16 == 0.
// Block: 256 threads = 8 waves; block covers 16 rows x 256 cols; each wave owns 16x32.
__global__ void __launch_bounds__(256)
gemm_wmma_f32_kernel(const float* __restrict__ X,
                     const float4* __restrict__ Bq,
                     const float* __restrict__ bias,
                     float* __restrict__ Y) {
    __shared__ float As[16][H + 4];   // +4 pad: fragment reads hit distinct LDS banks

    const int m0 = blockIdx.x * 16;
    const float* Xt = X + (size_t)m0 * H;
    for (int i = threadIdx.x; i < 16 * H; i += 256)
        As[i >> 8][i & (H - 1)] = Xt[i];
    __syncthreads();

    const int lane = threadIdx.x & 31;
    const int wid  = threadIdx.x >> 5;
    const int half = lane >> 4;       // 0: lanes 0-15, 1: lanes 16-31
    const int lr   = lane & 15;
    const int n0   = wid * 32;

    v8f c0, c1;
    const float bv0 = bias[n0 + lr];
    const float bv1 = bias[n0 + 16 + lr];
#pragma unroll
    for (int v = 0; v < 8; ++v) { c0[v] = bv0; c1[v] = bv1; }

    // per-wave base into packed B; per-lane offset within a k-step
    const float4* Bw = Bq + (size_t)wid * 16 + lr;

    // A 16x4 f32 layout (§7.12.2): VGPR0 = {K, K+2}, VGPR1 = {K+1, K+3}
    for (int ks = 0; ks < H / 4; ++ks) {
        const int ka = 4 * ks + 2 * half;
        v2f a;
        a.x = As[lr][ka];
        a.y = As[lr][ka + 1];

        const float4 bv = Bw[(ks * 2 + half) * 128];   // one b128: both col-tiles
        v2f b0, b1;
        b0.x = bv.x; b0.y = bv.y;
        b1.x = bv.z; b1.y = bv.w;

        c0 = __builtin_amdgcn_wmma_f32_16x16x4_f32(false, a, false, b0,
                                                   (short)0, c0, false, false);
        c1 = __builtin_amdgcn_wmma_f32_16x16x4_f32(false, a, false, b1,
                                                   (short)0, c1, false, false);
    }

    float* Yt = Y + (size_t)m0 * H;
#pragma unroll
    for (int v = 0; v < 8; ++v) {
        const int row = v + 8 * half;  // C/D: VGPR v holds M=v (lanes 0-15), M=v+8 (16-31)
        Yt[row * H + n0 + lr]      = c0[v];
        Yt[row * H + n0 + 16 + lr] = c1[v];
    }
}

// -------- edge scatter: dst[didx[e]] += (wnum[sidx[e]]/wden[didx[e]]) * src[sidx[e]] ----
// one wave32 per edge; lanes split the 256-wide row (2 float4 each)
__global__ void __launch_bounds__(256)
scatter_edges_kernel(const float* __restrict__ src,
                     const int* __restrict__ sidx,
                     const int* __restrict__ didx,
                     const float* __restrict__ wnum,
                     const float* __restrict__ wden,
                     float* __restrict__ dst, int numE) {
    const int gw   = (blockIdx.x * blockDim.x + threadIdx.x) >> 5;
    const int lane = threadIdx.x & 31;
    if (gw >= numE) return;

    const int s = sidx[gw];
    const int d = didx[gw];
    const float w = wnum[s] / wden[d];

    const float4* srow = (const float4*)(src + (size_t)s * H);
    float* drow = dst + (size_t)d * H;
#pragma unroll
    for (int i = 0; i < 2; ++i) {
        const int idx = lane + i * 32;            // 64 float4 per row
        float4 v = srow[idx];
        unsafeAtomicAdd(&drow[idx * 4 + 0], w * v.x);
        unsafeAtomicAdd(&drow[idx * 4 + 1], w * v.y);
        unsafeAtomicAdd(&drow[idx * 4 + 2], w * v.z);
        unsafeAtomicAdd(&drow[idx * 4 + 3], w * v.w);
    }
}

__global__ void prelu_inplace_kernel(float* __restrict__ p,
                                     const float* __restrict__ alpha, long n) {
    long i = (long)blockIdx.x * blockDim.x + threadIdx.x;
    if (i < n) {
        float x = p[i];
        float a = alpha[0];
        p[i] = x > 0.f ? x : a * x;
    }
}

extern "C" void kernel_launch(void* const* d_in, const int* in_sizes, int n_in,
                              void* d_out, int out_size, void* d_ws, size_t ws_size,
                              hipStream_t stream) {
    const float* n_feat          = (const float*)d_in[0];
    /* d_in[1] he_feat is unused by the reference */
    const int*   node_idx        = (const int*)d_in[2];
    const int*   hedge_idx       = (const int*)d_in[3];
    const float* node_reg_weight = (const float*)d_in[4];
    const float* node_reg_sum    = (const float*)d_in[5];
    const float* hedge_reg_weight= (const float*)d_in[6];
    const float* hedge_reg_sum   = (const float*)d_in[7];
    const float* W_in            = (const float*)d_in[8];
    const float* b_in            = (const float*)d_in[9];
    const float* W_n2e           = (const float*)d_in[10];
    const float* b_n2e           = (const float*)d_in[11];
    const float* W_e2n           = (const float*)d_in[12];
    const float* b_e2n           = (const float*)d_in[13];
    const float* alpha           = (const float*)d_in[14];

    const int NN = in_sizes[4];   // 50000
    const int NE = in_sizes[6];   // 10000
    const int E  = in_sizes[2];   // 300000

    // workspace layout (floats; all chunks 1KB-aligned -> float4-safe)
    float* ws    = (float*)d_ws;
    float* Wc    = ws;                 ws += H * H;          // fused W_n2e@W_in (row-major n,k)
    float* bc    = ws;                 ws += H;              // fused bias
    float* pad   = ws;                 ws += H - (H % H);    // keep alignment trivially
    float4* Bq_n = (float4*)ws;        ws += H * H;          // packed fused weight
    float4* Bq_e = (float4*)ws;        ws += H * H;          // packed W_e2n
    float* Wh_n  = ws;                 ws += (size_t)NN * H; // node-side GEMM result
    float* Wh_e  = ws;                                       // hedge-side GEMM result
    (void)pad;

    float* out      = (float*)d_out;
    float* node_out = out;                    // [NN,H]  (h_node accumulates here)
    float* efeat    = out + (size_t)NN * H;   // [NE,H]  (h_hedge accumulates here)

    // 1) fold weights/bias; pack both B matrices into WMMA fragment order
    combine_weights_kernel<<<H, H, 0, stream>>>(W_in, W_n2e, Wc);
    combine_bias_kernel<<<1, H, 0, stream>>>(W_n2e, b_in, b_n2e, bc);
    pack_b_kernel<<<(H * H / 4) / 256, 256, 0, stream>>>(Wc, Bq_n);
    pack_b_kernel<<<(H * H / 4) / 256, 256, 0, stream>>>(W_e2n, Bq_e);

    // 2) zero both output accumulation regions
    const long total = (long)(NN + NE) * H;
    zero_f32_kernel<<<(int)((total + 255) / 256), 256, 0, stream>>>(out, total);

    // 3) Wh_n = n_feat @ Wc^T + bc   (fused input2hidden + n2e)
    gemm_wmma_f32_kernel<<<NN / 16, 256, 0, stream>>>(n_feat, Bq_n, bc, Wh_n);

    // 4) node -> hedge segment sum (atomics into efeat region), then PReLU
    scatter_edges_kernel<<<(E * 32 + 255) / 256, 256, 0, stream>>>(
        Wh_n, node_idx, hedge_idx, node_reg_weight, hedge_reg_sum, efeat, E);
    prelu_inplace_kernel<<<(int)(((long)NE * H + 255) / 256), 256, 0, stream>>>(
        efeat, alpha, (long)NE * H);

    // 5) Wh_e = efeat @ W_e2n^T + b_e2n
    gemm_wmma_f32_kernel<<<NE / 16, 256, 0, stream>>>(efeat, Bq_e, b_e2n, Wh_e);

    // 6) hedge -> node segment sum, then PReLU
    scatter_edges_kernel<<<(E * 32 + 255) / 256, 256, 0, stream>>>(
        Wh_e, hedge_idx, node_idx, hedge_reg_weight, node_reg_sum, node_out, E);
    prelu_inplace_kernel<<<(int)(((long)NN * H + 255) / 256), 256, 0, stream>>>(
        node_out, alpha, (long)NN * H);
}